// PolygonMatchingLoss_80264348828076
// MI455X (gfx1250) — compile-verified
//
#include <hip/hip_runtime.h>
#include <hip/hip_bf16.h>
#include <math.h>

typedef __attribute__((ext_vector_type(2))) float v2f;
typedef __attribute__((ext_vector_type(8))) float v8f;

#define NB 4
#define NP 64
#define NV 32
#define ND 256
#define MARGINF 0.1f

// ---------------------------------------------------------------------------
// Kernel A: costs[b,g,p] = min over valid rotations r of masked-L1 / (2*min_nv)
// One block per (b,g). 256 threads = 8 waves. WMMA f32 16x16x4 does the
// v-reduction (A = masked L1 chunk, B = ones -> D rows = sums over v-chunk).
// ---------------------------------------------------------------------------
__global__ __launch_bounds__(256) void poly_costs_kernel(
    const float* __restrict__ gt,    // (B,P,V,2)
    const float* __restrict__ pred,  // (B,P,V,2)
    const int*   __restrict__ mask,  // (B,P,V)
    const int*   __restrict__ cls,   // (B,P)
    float*       __restrict__ costs) // (B,P,P) in workspace
{
    const int b = blockIdx.x >> 6;
    const int g = blockIdx.x & 63;
    const int tid = threadIdx.x;

    __shared__ __align__(16) float s_pred[NP * NV * 2]; // 16 KB
    __shared__ __align__(16) float s_rot [NV * NV * 2]; //  8 KB (rot[r][v][xy])
    __shared__ __align__(16) float s_num [NV * NP];     //  8 KB (numerator[r][p])
    __shared__ int s_nv[NP];
    __shared__ int s_minnv[NP];

    // nv[p] = count of mask==0 over V
    if (tid < NP) {
        const int* mp = mask + (b * NP + tid) * NV;
        int c = 0;
        #pragma unroll
        for (int v = 0; v < NV; ++v) c += (mp[v] == 0);
        s_nv[tid] = c;
    }
    __syncthreads();

    const int nv_g   = s_nv[g];
    const int is_poly = (cls[b * NP + g] == 1);
    const int n_valid = is_poly ? nv_g : 2;

    if (tid < NP) s_minnv[tid] = (nv_g < s_nv[tid]) ? nv_g : s_nv[tid];

    // Stage pred[b] into LDS (4096 floats) via float4
    {
        const float4* src = (const float4*)(pred + (size_t)b * NP * NV * 2);
        float4* dst = (float4*)s_pred;
        for (int i = tid; i < NP * NV * 2 / 4; i += 256) dst[i] = src[i];
    }
    // Build rotated gt table rot[r][v][2]
    {
        const float* gg = gt + ((size_t)b * NP + g) * NV * 2;
        const int nvm = (nv_g > 1) ? nv_g : 1;
        for (int e = tid; e < NV * NV; e += 256) {
            const int r = e >> 5, v = e & 31;
            int vtx;
            if (is_poly) vtx = (r + v) % nvm;
            else         vtx = (r == 0) ? v : (NV - 1 - v); // r>=2 unused
            s_rot[e * 2 + 0] = gg[vtx * 2 + 0];
            s_rot[e * 2 + 1] = gg[vtx * 2 + 1];
        }
    }
    __syncthreads();

    const int wave = tid >> 5;
    const int lane = tid & 31;
    const int half = lane >> 4;   // 0: K=0,1   1: K=2,3
    const int mrow = lane & 15;   // A-matrix row owned by this lane

    const v2f bones = {1.0f, 1.0f}; // B = all ones (layout independent)

    #pragma unroll
    for (int rr = 0; rr < 4; ++rr) {
        const int r = wave * 4 + rr;       // wave-uniform
        if (r < n_valid) {
            #pragma unroll
            for (int pt = 0; pt < 4; ++pt) {
                const int p = pt * 16 + mrow;
                const int mn = s_minnv[p];
                v8f acc = {};
                #pragma unroll
                for (int k = 0; k < 8; ++k) {
                    const int v = k * 4 + half * 2;    // this lane's K pair -> v, v+1
                    const float4 gv = *(const float4*)&s_rot [(r * NV + v) * 2];
                    const float4 pv = *(const float4*)&s_pred[(p * NV + v) * 2];
                    float a0 = fabsf(gv.x - pv.x) + fabsf(gv.y - pv.y);
                    float a1 = fabsf(gv.z - pv.z) + fabsf(gv.w - pv.w);
                    a0 = (v     < mn) ? a0 : 0.0f;
                    a1 = (v + 1 < mn) ? a1 : 0.0f;
                    v2f A = {a0, a1};
                    acc = __builtin_amdgcn_wmma_f32_16x16x4_f32(
                        false, A, false, bones, (short)0, acc, false, false);
                }
                // D[m][n] identical over n. Lane 0 holds rows 0-7 in acc[0..7],
                // lane 16 holds rows 8-15.
                if ((lane & 15) == 0) {
                    const int pb = pt * 16 + half * 8;
                    #pragma unroll
                    for (int j = 0; j < 8; ++j) s_num[r * NP + pb + j] = acc[j];
                }
            }
        }
    }
    __syncthreads();

    if (tid < NP) {
        float mv = INFINITY;
        for (int r = 0; r < n_valid; ++r) mv = fminf(mv, s_num[r * NP + tid]);
        const float denom = 2.0f * (float)s_minnv[tid]; // 0/0->NaN, inf/0->inf (matches ref)
        costs[((size_t)b * NP + g) * NP + tid] = mv / denom;
    }
}

// ---------------------------------------------------------------------------
// Kernel B: per-batch reductions. One block per b.
//  - perm / broken from the 64x64 costs matrix
//  - reg = mean_p sum_d mu^2 via WMMA f32 16x16x4 ones-trick (waves 0-3)
//  - sig = mean 1/sigma (waves 4-7)
// Deterministic: only int atomicOr; float sums reduced serially by thread 0.
// ---------------------------------------------------------------------------
__global__ __launch_bounds__(256) void poly_finalize_kernel(
    const float* __restrict__ costs, // (B,P,P)
    const float* __restrict__ mu,    // (B,P,D)
    const float* __restrict__ sigma, // (B,P,D)
    float*       __restrict__ out)   // 20 floats: batch|perm|reg|sig|broken
{
    const int b = blockIdx.x;
    const int tid = threadIdx.x;

    __shared__ __align__(16) float sc[NP * NP]; // 16 KB
    __shared__ float sd[NP];
    __shared__ float s_rowcol[NP];
    __shared__ float s_reg[8];
    __shared__ float s_sig[128];
    __shared__ int   s_broken;

    const float* cb = costs + (size_t)b * NP * NP;
    for (int i = tid; i < NP * NP; i += 256) sc[i] = cb[i];
    if (tid == 0) s_broken = 0;
    __syncthreads();
    if (tid < NP) sd[tid] = sc[tid * NP + tid];
    __syncthreads();

    if (tid < NP) {
        const int i = tid;
        const float di = sd[i];
        float rmax = 0.0f, cmax = 0.0f;
        int br = 0;
        for (int j = 0; j < NP; ++j) {
            if (j == i) continue;
            const float cij = sc[i * NP + j];
            const float cji = sc[j * NP + i];
            rmax = fmaxf(rmax, fmaxf(MARGINF - cij + di, 0.0f)); // cr row-max (row i)
            cmax = fmaxf(cmax, fmaxf(MARGINF - cji + di, 0.0f)); // cc col-max (col i)
            br |= ((cij < di) && (cji < sd[j])) ? 1 : 0;          // ar & ar^T
        }
        s_rowcol[i] = rmax + cmax;
        if (br) atomicOr(&s_broken, 1);
    }

    const int wave = tid >> 5;
    const int lane = tid & 31;
    const int half = lane >> 4;
    const int mrow = lane & 15;

    if (wave < 4) {
        // reg: rows = 16 p's of tile `wave`; A[m,k] = mu^2; B = ones
        const int p = wave * 16 + mrow;
        const float* mp = mu + ((size_t)b * NP + p) * ND;
        const v2f bones = {1.0f, 1.0f};
        v8f acc = {};
        #pragma unroll 8
        for (int k = 0; k < ND / 4; ++k) {
            const int d0 = k * 4 + half * 2;
            const float2 mv = *(const float2*)&mp[d0];
            v2f A = {mv.x * mv.x, mv.y * mv.y};
            acc = __builtin_amdgcn_wmma_f32_16x16x4_f32(
                false, A, false, bones, (short)0, acc, false, false);
        }
        if ((lane & 15) == 0) {
            float s = 0.0f;
            #pragma unroll
            for (int j = 0; j < 8; ++j) s += acc[j];
            s_reg[wave * 2 + half] = s;
        }
    } else {
        // sig: 128 threads cover 64*256 reciprocals
        const float* sg = sigma + (size_t)b * NP * ND;
        const int t = tid - 128;
        float s = 0.0f;
        for (int i = t; i < NP * ND; i += 128) s += 1.0f / sg[i];
        s_sig[t] = s;
    }
    __syncthreads();

    if (tid == 0) {
        float perm = 0.0f;
        for (int i = 0; i < NP; ++i) perm += s_rowcol[i];
        float reg = 0.0f;
        for (int i = 0; i < 8; ++i) reg += s_reg[i];
        reg *= (1.0f / 64.0f);
        float sig = 0.0f;
        for (int i = 0; i < 128; ++i) sig += s_sig[i];
        sig *= (1.0f / (64.0f * 256.0f));
        const float batch = 0.1f * reg + 0.05f * sig + 1.0f * perm;
        out[b]      = batch;
        out[4 + b]  = perm;
        out[8 + b]  = reg;
        out[12 + b] = sig;
        out[16 + b] = s_broken ? 1.0f : 0.0f;
    }
}

extern "C" void kernel_launch(void* const* d_in, const int* in_sizes, int n_in,
                              void* d_out, int out_size, void* d_ws, size_t ws_size,
                              hipStream_t stream) {
    const float* gt    = (const float*)d_in[0]; // all_gt_polys   (4,64,32,2)
    const float* pred  = (const float*)d_in[1]; // all_pred_polys (4,64,32,2)
    const float* mu    = (const float*)d_in[2]; // mu_pred        (4,64,256)
    const float* sigma = (const float*)d_in[3]; // sigma_pred     (4,64,256)
    const int*   mask  = (const int*)d_in[4];   // all_poly_mask  (4,64,32)
    const int*   cls   = (const int*)d_in[5];   // all_poly_class (4,64)
    float* out   = (float*)d_out;               // 20 floats
    float* costs = (float*)d_ws;                // 4*64*64 floats = 64 KB scratch

    poly_costs_kernel<<<NB * NP, 256, 0, stream>>>(gt, pred, mask, cls, costs);
    poly_finalize_kernel<<<NB, 256, 0, stream>>>(costs, mu, sigma, out);
}